// BallQueryAttention_57715770523981
// MI455X (gfx1250) — compile-verified
//
#include <hip/hip_runtime.h>

// BallQueryAttention for MI455X (gfx1250, wave32).
// out = softmax(1[dist<=R]) @ x  ==  ( (e-1)*(mask@x) + colsum(x) ) / (N + (e-1)*cnt)
// mask_ij = (2*G_ij >= sq_i + sq_j - R^2),  G = x @ x^T  (fp32 WMMA for threshold precision)

typedef __attribute__((ext_vector_type(2))) float v2f;
typedef __attribute__((ext_vector_type(8))) float v8f;

#define N_PTS 8192
#define DIM   64
#define R2    121.0f                    // RADIUS^2
#define EM1   1.7182818284590452354f    // e - 1

// ---------------- prologue 1: sq[i] = sum_d x[i][d]^2 ----------------
__global__ void bq_rowsq(const float* __restrict__ x, float* __restrict__ sq) {
    int i = blockIdx.x * blockDim.x + threadIdx.x;
    const float4* xr = (const float4*)(x + i * DIM);
    float s = 0.f;
#pragma unroll
    for (int k = 0; k < DIM / 4; ++k) {
        float4 v = xr[k];
        s += v.x * v.x + v.y * v.y + v.z * v.z + v.w * v.w;
    }
    sq[i] = s;
}

// ---------------- prologue 2: stot[d] = sum_i x[i][d] ----------------
__global__ void bq_colsum(const float* __restrict__ x, float* __restrict__ stot) {
    __shared__ float part[4][DIM];
    int d = threadIdx.x & 63;
    int g = threadIdx.x >> 6;
    float s = 0.f;
    for (int i = g; i < N_PTS; i += 4) s += x[i * DIM + d];
    part[g][d] = s;
    __syncthreads();
    if (threadIdx.x < DIM)
        stot[d] = part[0][d] + part[1][d] + part[2][d] + part[3][d];
}

// ---------------- main kernel ----------------
// Block = 128 threads = 4 waves; wave w owns i-tile [ib, ib+16).
// j-tiles of 32 rows are staged to LDS per barrier and consumed as two
// 16-row subtiles; all WMMA operand fragments are prefetched per subtile.
#define LDS_STRIDE 68   // 64 + 4 pad: conflict-free banks, 16B-aligned rows

__launch_bounds__(128)
__global__ void bq_attn(const float* __restrict__ x,
                        const float* __restrict__ sq,
                        const float* __restrict__ stot,
                        float* __restrict__ out) {
    __shared__ __align__(16) float ldsx[32 * LDS_STRIDE];
    __shared__ float ldssq[32];

    const int tid   = threadIdx.x;
    const int wave  = tid >> 5;
    const int lane  = tid & 31;
    const int lan16 = lane & 15;
    const bool hi   = (lane & 16) != 0;     // upper half of wave
    const int  h2   = hi ? 2 : 0;           // K-offset contributed by lane half

    const int ib = blockIdx.x * 64 + wave * 16;

    // Hoisted B fragments of the i-tile for the Gram matmul:
    // bi[dc] @ lane l = { x[ib+l%16][4dc + 2h], x[ib+l%16][4dc+1 + 2h] }
    v2f bi[16];
    {
        const float* xi = x + (ib + lan16) * DIM + h2;
#pragma unroll
        for (int dc = 0; dc < 16; ++dc) {
            bi[dc].x = xi[4 * dc + 0];
            bi[dc].y = xi[4 * dc + 1];
        }
    }
    const float si = sq[ib + lan16];

    // O^T accumulators: acc[m] reg r lane l = Sball[ ib+l%16 ][ 16m + r + 8h ]
    v8f acc[4];
#pragma unroll
    for (int m = 0; m < 4; ++m) acc[m] = (v8f){0.f,0.f,0.f,0.f,0.f,0.f,0.f,0.f};
    float cnt = 0.f;   // per-lane: ball count for i = ib+l%16 over this half's j's

    for (int jb = 0; jb < N_PTS; jb += 32) {
        __syncthreads();   // previous tile fully consumed
        {   // stage x[jb..jb+31][0..63] -> LDS (padded rows), plus sq[jb..jb+31]
            int row = tid >> 2;            // 0..31
            int col = (tid & 3) * 16;      // 0, 16, 32, 48
            const float4* src = (const float4*)(x + (jb + row) * DIM + col);
            float4 a = src[0], b = src[1], c = src[2], d = src[3];
            float4* dst = (float4*)(ldsx + row * LDS_STRIDE + col);
            dst[0] = a; dst[1] = b; dst[2] = c; dst[3] = d;
            if (tid < 32) ldssq[tid] = sq[jb + tid];
        }
        __syncthreads();

#pragma unroll
        for (int sub = 0; sub < 2; ++sub) {
            const int rbase = sub * 16;    // subtile row base inside LDS stage

            // ---- prefetch Gram A fragments:
            // af[dc] lane l = { x[jb+rbase+l%16][4dc+2h], ...[4dc+1+2h] }
            v2f af[16];
            {
                const float* ap = ldsx + (rbase + lan16) * LDS_STRIDE + h2;
#pragma unroll
                for (int dc = 0; dc < 16; ++dc) {
                    af[dc].x = ap[4 * dc + 0];
                    af[dc].y = ap[4 * dc + 1];
                }
            }
            // ---- prefetch O^T A fragments:
            // aot[kc][mm] lane l = { x[jb+rbase+4kc+2h][16mm+l%16],
            //                        x[jb+rbase+4kc+1+2h][16mm+l%16] }
            v2f aot[4][4];
            {
#pragma unroll
                for (int kc = 0; kc < 4; ++kc) {
                    const float* ap0 = ldsx + (rbase + 4 * kc + 0 + h2) * LDS_STRIDE + lan16;
                    const float* ap1 = ldsx + (rbase + 4 * kc + 1 + h2) * LDS_STRIDE + lan16;
#pragma unroll
                    for (int mm = 0; mm < 4; ++mm) {
                        aot[kc][mm].x = ap0[16 * mm];
                        aot[kc][mm].y = ap1[16 * mm];
                    }
                }
            }

            // ---- Gram^T tile: g reg r lane l = dot(x[jb+rbase+r+8h], x[ib+l%16])
            v8f g = (v8f){0.f,0.f,0.f,0.f,0.f,0.f,0.f,0.f};
#pragma unroll
            for (int dc = 0; dc < 16; ++dc)
                g = __builtin_amdgcn_wmma_f32_16x16x4_f32(false, af[dc], false, bi[dc],
                                                          (short)0, g, false, false);

            // ---- mask (D-layout) + count ----
            float m[8];
#pragma unroll
            for (int r = 0; r < 8; ++r) {
                float sj = ldssq[rbase + r + (hi ? 8 : 0)];
                m[r] = (2.0f * g[r] + R2 >= si + sj) ? 1.0f : 0.0f;
                cnt += m[r];
            }

            // cross-half copies: ms[r] lane l = m[r] lane l^16 (ds_swizzle SWAPX16)
            float ms[8];
#pragma unroll
            for (int r = 0; r < 8; ++r)
                ms[r] = __int_as_float(
                    __builtin_amdgcn_ds_swizzle(__float_as_int(m[r]), 0x401f));

            // Re-layout mask D-fragment -> B-fragments bm[kc] (K=j, N=i):
            // bm[kc][v] lane l holds mask(i=l%16, jloc = 4kc+v+2h)
            v2f bm[4];
            bm[0].x = hi ? ms[2] : m[0];  bm[0].y = hi ? ms[3] : m[1];
            bm[1].x = hi ? ms[6] : m[4];  bm[1].y = hi ? ms[7] : m[5];
            bm[2].x = hi ? m[2] : ms[0];  bm[2].y = hi ? m[3] : ms[1];
            bm[3].x = hi ? m[6] : ms[4];  bm[3].y = hi ? m[7] : ms[5];

            // ---- O^T += x^T(d,j) @ mask^T(j,i) ----
#pragma unroll
            for (int kc = 0; kc < 4; ++kc)
#pragma unroll
                for (int mm = 0; mm < 4; ++mm)
                    acc[mm] = __builtin_amdgcn_wmma_f32_16x16x4_f32(
                        false, aot[kc][mm], false, bm[kc], (short)0, acc[mm],
                        false, false);
        }
    }

    // combine the two lane-halves' partial counts
    float cntFull = cnt + __int_as_float(
        __builtin_amdgcn_ds_swizzle(__float_as_int(cnt), 0x401f));
    float inv = 1.0f / ((float)N_PTS + EM1 * cntFull);

    // epilogue: out[i][d] = (EM1 * Sball + stot[d]) * inv
    const int iOut = ib + lan16;
#pragma unroll
    for (int mm = 0; mm < 4; ++mm) {
#pragma unroll
        for (int r = 0; r < 8; ++r) {
            int d = 16 * mm + r + (hi ? 8 : 0);
            out[iOut * DIM + d] = (EM1 * acc[mm][r] + stot[d]) * inv;
        }
    }
}

// ---------------- launcher ----------------
extern "C" void kernel_launch(void* const* d_in, const int* in_sizes, int n_in,
                              void* d_out, int out_size, void* d_ws, size_t ws_size,
                              hipStream_t stream) {
    const float* x = (const float*)d_in[0];   // [8192, 64] fp32 (adj = d_in[1] unused)
    float* out  = (float*)d_out;              // [8192, 64] fp32
    float* sq   = (float*)d_ws;               // 8192 floats
    float* stot = sq + N_PTS;                 // 64 floats

    bq_rowsq <<<N_PTS / 256, 256, 0, stream>>>(x, sq);
    bq_colsum<<<1,           256, 0, stream>>>(x, stot);
    bq_attn  <<<N_PTS / 64,  128, 0, stream>>>(x, sq, stot, out);
}